// Graph2GraphModel_36893769072882
// MI455X (gfx1250) — compile-verified
//
#include <hip/hip_runtime.h>
#include <hip/hip_bf16.h>
#include <math.h>

#define NS   360   // scan points
#define NP   368   // padded to multiple of 16 for WMMA tiles
#define HID  64

typedef __attribute__((ext_vector_type(2))) float v2f;
typedef __attribute__((ext_vector_type(8))) float v8f;

// ---------------------------------------------------------------------------
// Kernel 1: everything up to c = gemb @ Wp + bp, fully LDS-resident.
// One workgroup (16 wave32s). GCN GEMMs use V_WMMA_F32_16X16X4_F32.
// ---------------------------------------------------------------------------
__global__ __launch_bounds__(512)
void gcn_fused_kernel(const float* __restrict__ x,
                      const float* __restrict__ W1, const float* __restrict__ b1,
                      const float* __restrict__ W2, const float* __restrict__ b2,
                      const float* __restrict__ W3, const float* __restrict__ b3,
                      const float* __restrict__ Wp, const float* __restrict__ bp,
                      float* __restrict__ c_out)
{
    __shared__ float hA[NP * HID];     // layer activations (padded rows zero)
    __shared__ float hB[NP * HID];     // h @ W scratch
    __shared__ float Wl[HID * HID];    // current layer weight
    __shared__ float n0s[NP], n1s[NP], invs[NP], inv2s[NP], coefs[NP], masks[NP];
    __shared__ float red[8 * HID];
    __shared__ float gemb[HID];
    __shared__ float s_nvalid;

    const int tid  = threadIdx.x;
    const int wave = tid >> 5;
    const int lane = tid & 31;

    if (tid == 0) s_nvalid = 0.0f;

    // ---- preprocessing: mask, polar nodes ----
    if (tid < NP) {
        float m = 0.0f, a0 = 0.0f, a1 = 0.0f;
        if (tid < NS) {
            float lid = x[tid];                       // x[0, tid]
            m  = (lid != 1.0f) ? 1.0f : 0.0f;
            float ang = (float)tid * (6.28318530717958647692f / 359.0f);
            a0 = lid * cosf(ang);
            a1 = lid * sinf(ang);
        }
        masks[tid] = m; n0s[tid] = a0; n1s[tid] = a1;
    }
    __syncthreads();

    // ---- deg -> inv, inv^2 ----
    if (tid < NP) {
        float iv = 0.0f;
        if (tid < NS) {
            float m  = masks[tid];
            float pl = (tid > 0)      ? masks[tid - 1] * m : 0.0f;
            float pr = (tid < NS - 1) ? m * masks[tid + 1] : 0.0f;
            float deg = m + pl + pr;
            iv = (deg > 0.0f) ? 1.0f / sqrtf(deg) : 0.0f;
        }
        invs[tid] = iv;
        inv2s[tid] = iv * iv;
    }
    if (tid < NS) atomicAdd(&s_nvalid, masks[tid]);
    __syncthreads();

    // ---- edge coefficients ----
    if (tid < NP) {
        float cf = 0.0f;
        if (tid < NS - 1) cf = masks[tid] * masks[tid + 1] * invs[tid] * invs[tid + 1];
        coefs[tid] = cf;
    }
    __syncthreads();

    // ---- tridiagonal aggregate + bias + relu : src -> dst ----
    auto aggregate = [&](const float* src, float* dst, const float* b) {
        for (int idx = tid; idx < NP * HID; idx += 512) {
            int i = idx >> 6, j = idx & 63;
            float v = 0.0f;
            if (i < NS) {
                v = inv2s[i] * src[idx];
                if (i > 0) v += coefs[i - 1] * src[idx - HID];
                v += coefs[i] * src[idx + HID];   // coefs[NS-1]==0, row NS is zero pad
                v += b[j];
                v = fmaxf(v, 0.0f);
            }
            dst[idx] = v;                          // pad rows stay exactly zero
        }
    };

    // ---- GEMM: dst(NP x 64) = src(NP x 64) @ Wl(64 x 64) via f32 WMMA ----
    auto gemm = [&](const float* src, float* dst) {
        const int row  = lane & 15;
        const int kOff = (lane >> 4) << 1;         // lanes 0-15 -> K 0,1 ; lanes 16-31 -> K 2,3
        for (int t = wave; t < (NP / 16) * (HID / 16); t += 16) {   // 23 * 4 = 92 tiles
            const int m0 = (t >> 2) * 16;
            const int n0 = (t & 3) * 16;
            v8f acc = {0.0f, 0.0f, 0.0f, 0.0f, 0.0f, 0.0f, 0.0f, 0.0f};
#pragma unroll
            for (int k = 0; k < HID; k += 4) {
                const float* ap = &src[(m0 + row) * HID + k + kOff];
                v2f a; a.x = ap[0]; a.y = ap[1];
                const float* bq = &Wl[(k + kOff) * HID + n0 + row];
                v2f b; b.x = bq[0]; b.y = bq[HID];
                acc = __builtin_amdgcn_wmma_f32_16x16x4_f32(
                        false, a, false, b, (short)0, acc, false, false);
            }
            const int rbase = m0 + ((lane >> 4) << 3);   // +8 for upper half-wave
#pragma unroll
            for (int v = 0; v < 8; ++v)
                dst[(rbase + v) * HID + n0 + row] = acc[v];
        }
    };

    // ---- layer 1 (K=2, plain FMA) ----
    for (int idx = tid; idx < NP * HID; idx += 512) {
        int i = idx >> 6, j = idx & 63;
        float v = 0.0f;
        if (i < NS) v = n0s[i] * W1[j] + n1s[i] * W1[HID + j];
        hB[idx] = v;
    }
    __syncthreads();
    aggregate(hB, hA, b1);
    __syncthreads();

    // ---- layer 2 ----
    for (int idx = tid; idx < HID * HID; idx += 512) Wl[idx] = W2[idx];
    __syncthreads();
    gemm(hA, hB);
    __syncthreads();
    aggregate(hB, hA, b2);
    __syncthreads();

    // ---- layer 3 ----
    for (int idx = tid; idx < HID * HID; idx += 512) Wl[idx] = W3[idx];
    __syncthreads();
    gemm(hA, hB);
    __syncthreads();
    aggregate(hB, hA, b3);
    __syncthreads();

    // ---- masked mean pool ----
    {
        int j = tid & 63, part = tid >> 6;        // 8 partial sums per column
        float s = 0.0f;
        for (int i = part * 45; i < (part + 1) * 45; ++i)
            s += hA[i * HID + j] * masks[i];
        red[part * HID + j] = s;
    }
    __syncthreads();
    if (tid < HID) {
        float s = 0.0f;
#pragma unroll
        for (int p = 0; p < 8; ++p) s += red[p * HID + tid];
        gemb[tid] = s / s_nvalid;
    }
    __syncthreads();

    // ---- c = gemb @ Wp + bp  (64 -> 512), one output per thread ----
    {
        float acc = bp[tid];
#pragma unroll 8
        for (int k = 0; k < HID; ++k) acc += gemb[k] * Wp[k * 512 + tid];
        c_out[tid] = acc;
    }
}

// ---------------------------------------------------------------------------
// Kernel 2: t = relu(c @ Wm1 + bm1)   (512 -> 1024), 4 WGs read 2MB in parallel
// ---------------------------------------------------------------------------
__global__ __launch_bounds__(256)
void head1_kernel(const float* __restrict__ c, const float* __restrict__ Wm1,
                  const float* __restrict__ bm1, float* __restrict__ t)
{
    __shared__ float cs[512];
    const int tid = threadIdx.x;
    cs[tid]       = c[tid];
    cs[tid + 256] = c[tid + 256];
    __syncthreads();
    const int j = blockIdx.x * 256 + tid;
    float acc = bm1[j];
#pragma unroll 8
    for (int k = 0; k < 512; ++k) acc += cs[k] * Wm1[k * 1024 + j];
    t[j] = fmaxf(acc, 0.0f);
}

// ---------------------------------------------------------------------------
// Kernel 3: out = t @ Wm2 + bm2       (1024 -> 200)
// ---------------------------------------------------------------------------
__global__ __launch_bounds__(256)
void head2_kernel(const float* __restrict__ t, const float* __restrict__ Wm2,
                  const float* __restrict__ bm2, float* __restrict__ out)
{
    __shared__ float ts[1024];
    const int tid = threadIdx.x;
    for (int k = tid; k < 1024; k += 256) ts[k] = t[k];
    __syncthreads();
    if (tid < 200) {
        float acc = bm2[tid];
#pragma unroll 8
        for (int k = 0; k < 1024; ++k) acc += ts[k] * Wm2[k * 200 + tid];
        out[tid] = acc;
    }
}

// ---------------------------------------------------------------------------
extern "C" void kernel_launch(void* const* d_in, const int* in_sizes, int n_in,
                              void* d_out, int out_size, void* d_ws, size_t ws_size,
                              hipStream_t stream)
{
    const float* x   = (const float*)d_in[0];
    const float* W1  = (const float*)d_in[1];
    const float* b1  = (const float*)d_in[2];
    const float* W2  = (const float*)d_in[3];
    const float* b2  = (const float*)d_in[4];
    const float* W3  = (const float*)d_in[5];
    const float* b3  = (const float*)d_in[6];
    const float* Wp  = (const float*)d_in[7];
    const float* bp  = (const float*)d_in[8];
    const float* Wm1 = (const float*)d_in[9];
    const float* bm1 = (const float*)d_in[10];
    const float* Wm2 = (const float*)d_in[11];
    const float* bm2 = (const float*)d_in[12];

    float* ws = (float*)d_ws;
    float* c  = ws;          // 512 floats
    float* t  = ws + 512;    // 1024 floats

    gcn_fused_kernel<<<1, 512, 0, stream>>>(x, W1, b1, W2, b2, W3, b3, Wp, bp, c);
    head1_kernel<<<4, 256, 0, stream>>>(c, Wm1, bm1, t);
    head2_kernel<<<1, 256, 0, stream>>>(t, Wm2, bm2, (float*)d_out);
}